// EfficientMambaAttention_65025804861605
// MI455X (gfx1250) — compile-verified
//
#include <hip/hip_runtime.h>
#include <hip/hip_bf16.h>

// EfficientMambaAttention for MI455X (gfx1250, wave32).
// Pipeline: means -> [in_proj WMMA] -> conv+silu -> [x_proj WMMA] -> scan ->
//           [out_proj WMMA + sigmoid] (x2 paths) -> gated elementwise output.
// Memory-bound overall (~600 MB HBM traffic -> ~26us floor at 23.3 TB/s);
// the tiny f32 GEMMs ride V_WMMA_F32_16X16X4_F32.

#define BATCH   8
#define DMODEL  96
#define HH      256
#define WW      256
#define DIN     192
#define DTR     6
#define DST     16
#define ROWS    (BATCH * HH)        // 2048 rows for both paths (L=256)
#define XZN     (2 * DIN)           // 384
#define XDBL_STRIDE 48              // 38 padded to 3 WMMA tiles

typedef __attribute__((ext_vector_type(2))) float v2f;
typedef __attribute__((ext_vector_type(8))) float v8f;

__device__ __forceinline__ float sigmoidf_fast(float v) {
    return 1.0f / (1.0f + __expf(-v));
}

// ---------------------------------------------------------------------------
// Kernel A: per (b,c) plane, produce row means (ux[b,h,c]) and col means
// (uy[b,w,c]) in one kernel. Pass 1 coalesced over w; pass 2 re-reads the
// 256KB plane (L2-resident) with float4 per-thread rows.
// ---------------------------------------------------------------------------
__global__ __launch_bounds__(256)
void means_kernel(const float* __restrict__ x, float* __restrict__ ux,
                  float* __restrict__ uy) {
    const int bc = blockIdx.x;              // 0..767
    const int b = bc / DMODEL, c = bc % DMODEL;
    const int t = threadIdx.x;              // 0..255
    const float* plane = x + (size_t)bc * (HH * WW);

    // column sums over h -> uy[b, w=t, c]
    float cs = 0.0f;
    for (int h = 0; h < HH; ++h) cs += plane[h * WW + t];
    uy[(b * WW + t) * DMODEL + c] = cs * (1.0f / (float)HH);

    // row sums over w -> ux[b, h=t, c]
    const float4* row = (const float4*)(plane + t * WW);
    float rs = 0.0f;
    for (int w4 = 0; w4 < WW / 4; ++w4) {
        float4 v = row[w4];
        rs += v.x + v.y + v.z + v.w;
    }
    ux[(b * HH + t) * DMODEL + c] = rs * (1.0f / (float)WW);
}

// ---------------------------------------------------------------------------
// Generic f32 WMMA GEMM: Cout[M x Nstride] = A[M x K] * Wt[Nreal x K]^T
// One wave per 16x16 tile; grids are sized so every wave owns a full tile
// (EXEC all-ones for every executing WMMA, ISA 7.12 requirement).
// B-side out-of-range lanes are handled branchlessly: the address is clamped
// to a valid row and the loaded fragment is zeroed with a select, so the
// inner loop stays a pair of global_load_b64 + v_wmma (no EXEC churn).
// act: 0 = identity, 1 = sigmoid (fused for the gate projection).
// A-fragment layout (32-bit A 16x4): lanes 0-15 K={k0,k0+1}, lanes 16-31
// K={k0+2,k0+3}; B mirrors (4x16, N striped across lane&15).
// ---------------------------------------------------------------------------
__global__ __launch_bounds__(256)
void gemm_wmma_f32(const float* __restrict__ A, const float* __restrict__ Wt,
                   float* __restrict__ Cout, int K, int Nreal, int Nstride,
                   int ntiles, int act) {
    const int wave = threadIdx.x >> 5;
    const int lane = threadIdx.x & 31;
    const int tile = blockIdx.x * 8 + wave;
    const int mt = tile / ntiles;
    const int nt = tile % ntiles;
    const int half = lane >> 4;              // 0 or 1
    const int l16  = lane & 15;
    const int arow = mt * 16 + l16;
    const int bcol = nt * 16 + l16;
    const bool bvalid = (bcol < Nreal);
    const int brow = bvalid ? bcol : 0;      // clamp: load stays in-bounds

    const float2* Arow2 = (const float2*)(A  + arow * K + half * 2);
    const float2* Brow2 = (const float2*)(Wt + brow * K + half * 2);

    v8f acc = {};
#pragma unroll 4
    for (int k0 = 0; k0 < K; k0 += 4) {
        const float2 av = Arow2[k0 >> 1];    // unconditional b64 load
        const float2 bv = Brow2[k0 >> 1];    // unconditional b64 load
        v2f a, bfrag;
        a.x = av.x;
        a.y = av.y;
        bfrag.x = bvalid ? bv.x : 0.0f;      // v_cndmask, no branch
        bfrag.y = bvalid ? bv.y : 0.0f;
        acc = __builtin_amdgcn_wmma_f32_16x16x4_f32(
            /*neg_a=*/false, a, /*neg_b=*/false, bfrag,
            /*c_mod=*/(short)0, acc, /*reuse_a=*/false, /*reuse_b=*/false);
    }

#pragma unroll
    for (int v = 0; v < 8; ++v) {
        const int r = mt * 16 + v + half * 8;
        if (bvalid) {
            float val = acc[v];
            if (act == 1) val = sigmoidf_fast(val);
            Cout[r * Nstride + bcol] = val;
        }
    }
}

// ---------------------------------------------------------------------------
// Causal depthwise conv (D_CONV=4, left pad 3) + bias + SiLU on the xs half
// of xz.  One thread per (row, d).
// ---------------------------------------------------------------------------
__global__ __launch_bounds__(256)
void conv_silu_kernel(const float* __restrict__ xz,
                      const float* __restrict__ conv_w,
                      const float* __restrict__ conv_b,
                      float* __restrict__ xs_act) {
    const int tid = blockIdx.x * 256 + threadIdx.x;   // < ROWS*DIN
    const int d = tid % DIN;
    const int row = tid / DIN;
    const int l = row & (HH - 1);
    float acc = conv_b[d];
#pragma unroll
    for (int k = 0; k < 4; ++k) {
        const int ls = l - 3 + k;
        if (ls >= 0) acc += conv_w[d * 4 + k] * xz[(row - 3 + k) * XZN + d];
    }
    xs_act[row * DIN + d] = acc * sigmoidf_fast(acc);   // SiLU
}

// ---------------------------------------------------------------------------
// Selective scan: 1536 independent (b, d) recurrences, h[16] in registers.
// dt = softplus(x_dbl[:,0:6] @ dt_w[d] + dt_b[d]); dA = exp(dt * -exp(Alog));
// h = dA*h + dt*B*xs; y = sum(h*C); out = (y + xs*D) * silu(z).
// exp/log go to the transcendental pipe and co-execute with the VALU FMAs.
// ---------------------------------------------------------------------------
__global__ __launch_bounds__(256)
void scan_kernel(const float* __restrict__ xdbl, const float* __restrict__ xs_act,
                 const float* __restrict__ xz, const float* __restrict__ dt_w,
                 const float* __restrict__ dt_b, const float* __restrict__ Alog,
                 const float* __restrict__ Dp, float* __restrict__ y_act) {
    const int tid = blockIdx.x * 256 + threadIdx.x;   // < BATCH*DIN = 1536
    const int d = tid % DIN;
    const int b = tid / DIN;

    float a_s[DST], h[DST];
#pragma unroll
    for (int s = 0; s < DST; ++s) {
        a_s[s] = -__expf(Alog[d * DST + s]);
        h[s] = 0.0f;
    }
    float w_dt[DTR];
#pragma unroll
    for (int r = 0; r < DTR; ++r) w_dt[r] = dt_w[d * DTR + r];
    const float bias = dt_b[d];
    const float Dd = Dp[d];

    for (int l = 0; l < HH; ++l) {
        const int row = b * HH + l;
        const float* xd = xdbl + row * XDBL_STRIDE;
        float dtr = bias;
#pragma unroll
        for (int r = 0; r < DTR; ++r) dtr += w_dt[r] * xd[r];
        const float dt = (dtr > 20.0f) ? dtr : __logf(1.0f + __expf(dtr));
        const float xs = xs_act[row * DIN + d];
        float y = 0.0f;
#pragma unroll
        for (int s = 0; s < DST; ++s) {
            const float dA = __expf(dt * a_s[s]);
            h[s] = dA * h[s] + (dt * xd[DTR + s]) * xs;
            y += h[s] * xd[DTR + DST + s];
        }
        const float zz = xz[row * XZN + DIN + d];
        y_act[row * DIN + d] = (y + xs * Dd) * (zz * sigmoidf_fast(zz));
    }
}

// ---------------------------------------------------------------------------
// Final gating: out[b,c,h,w] = x[b,c,h,w] * gx[b,h,c] * gy[b,w,c].
// One block per (b,c) plane; gy column staged once in LDS; x read/write
// fully coalesced (single HBM pass each).
// ---------------------------------------------------------------------------
__global__ __launch_bounds__(256)
void gate_mul_kernel(const float* __restrict__ x, const float* __restrict__ gx,
                     const float* __restrict__ gy, float* __restrict__ out) {
    const int bc = blockIdx.x;
    const int b = bc / DMODEL, c = bc % DMODEL;
    const int t = threadIdx.x;
    __shared__ float gyrow[WW];
    gyrow[t] = gy[(b * WW + t) * DMODEL + c];
    __syncthreads();
    const float* plane = x + (size_t)bc * (HH * WW);
    float* oplane = out + (size_t)bc * (HH * WW);
    const float gcol = gyrow[t];
    for (int h = 0; h < HH; ++h) {
        const float g = gx[(b * HH + h) * DMODEL + c];
        oplane[h * WW + t] = plane[h * WW + t] * g * gcol;
    }
}

// ---------------------------------------------------------------------------
extern "C" void kernel_launch(void* const* d_in, const int* in_sizes, int n_in,
                              void* d_out, int out_size, void* d_ws, size_t ws_size,
                              hipStream_t stream) {
    const float* x = (const float*)d_in[0];
    struct MP {
        const float *in_w, *conv_w, *conv_b, *xproj_w, *dt_w, *dt_b, *Alog, *D, *out_w;
    };
    MP mx = { (const float*)d_in[1], (const float*)d_in[2], (const float*)d_in[3],
              (const float*)d_in[4], (const float*)d_in[5], (const float*)d_in[6],
              (const float*)d_in[7], (const float*)d_in[8], (const float*)d_in[9] };
    MP my = { (const float*)d_in[10], (const float*)d_in[11], (const float*)d_in[12],
              (const float*)d_in[13], (const float*)d_in[14], (const float*)d_in[15],
              (const float*)d_in[16], (const float*)d_in[17], (const float*)d_in[18] };

    // Workspace layout (floats). Buffers reused across the two sequential
    // paths; only the gates persist. Total ~9.8 MB.
    float* ws = (float*)d_ws;
    float* ux    = ws; ws += ROWS * DMODEL;          // 196608
    float* uy    = ws; ws += ROWS * DMODEL;          // 196608
    float* xzbuf = ws; ws += ROWS * XZN;             // 786432
    float* xsbuf = ws; ws += ROWS * DIN;             // 393216
    float* xdbuf = ws; ws += ROWS * XDBL_STRIDE;     // 98304
    float* yabuf = ws; ws += ROWS * DIN;             // 393216
    float* gateX = ws; ws += ROWS * DMODEL;          // 196608
    float* gateY = ws; ws += ROWS * DMODEL;          // 196608

    // 1) pooled sequences
    means_kernel<<<BATCH * DMODEL, 256, 0, stream>>>(x, ux, uy);

    auto run_path = [&](const float* u, const MP& p, float* gate) {
        // in_proj: [2048,96] x [96,384]  -> tiles 128 x 24 = 3072 waves
        gemm_wmma_f32<<<(128 * 24) / 8, 256, 0, stream>>>(
            u, p.in_w, xzbuf, DMODEL, XZN, XZN, 24, 0);
        // depthwise conv + silu
        conv_silu_kernel<<<(ROWS * DIN) / 256, 256, 0, stream>>>(
            xzbuf, p.conv_w, p.conv_b, xsbuf);
        // x_proj: [2048,192] x [192,38] -> tiles 128 x 3 (N padded to 48)
        gemm_wmma_f32<<<(128 * 3) / 8, 256, 0, stream>>>(
            xsbuf, p.xproj_w, xdbuf, DIN, DTR + 2 * DST, XDBL_STRIDE, 3, 0);
        // selective scan (sequential in L, parallel in b*d)
        scan_kernel<<<(BATCH * DIN) / 256, 256, 0, stream>>>(
            xdbuf, xsbuf, xzbuf, p.dt_w, p.dt_b, p.Alog, p.D, yabuf);
        // out_proj + sigmoid: [2048,192] x [192,96] -> tiles 128 x 6
        gemm_wmma_f32<<<(128 * 6) / 8, 256, 0, stream>>>(
            yabuf, p.out_w, gate, DIN, DMODEL, DMODEL, 6, 1);
    };

    run_path(ux, mx, gateX);   // row path (sequence over H)
    run_path(uy, my, gateY);   // col path (sequence over W)

    // 2) fused gating, single read + write of x
    gate_mul_kernel<<<BATCH * DMODEL, 256, 0, stream>>>(x, gateX, gateY, (float*)d_out);
}